// AtomwiseLinear_TCSM_47304769798605
// MI455X (gfx1250) — compile-verified
//
#include <hip/hip_runtime.h>

typedef __attribute__((ext_vector_type(2))) float v2f;
typedef __attribute__((ext_vector_type(8))) float v8f;

#define DEG_THRESH 10
#define FD 30
#define BIGI 0x7fffffff

__global__ void init_ws_kernel(int* __restrict__ count, int* __restrict__ mix, int n) {
  int i = blockIdx.x * blockDim.x + threadIdx.x;
  if (i < n) { count[i] = 0; mix[i] = BIGI; }
}

__global__ void edge_kernel(const int* __restrict__ ei, const int* __restrict__ types,
                            int* __restrict__ count, int* __restrict__ mix, int ne) {
  int e = blockIdx.x * blockDim.x + threadIdx.x;
  if (e >= ne) return;
  // edge_index is streamed once: non-temporal so it doesn't evict count/mix from L2
  int s = __builtin_nontemporal_load(&ei[e]);        // edge_index[0][e]
  int d = __builtin_nontemporal_load(&ei[ne + e]);   // edge_index[1][e]
  atomicAdd(&count[s], 1);
  atomicAdd(&count[d], 1);
  int ts = types[s], td = types[d];
  int val = (ts == td) ? (ts + 1) : 0;
  atomicMin(&mix[s], val);
  atomicMin(&mix[d], val);
}

// One wave handles a 16-node x 32-feature tile (features padded 30->32).
// K (=30) padded to 32 -> 8 chunks of V_WMMA_F32_16X16X4_F32 per 16x16 output tile.
// A 16x4 f32 layout (ISA 7.12.2): lane half h=lane/16 selects K-pair, vgpr v selects lsb:
//   K = 4*c + 2*h + v, M = lane%16.   B 4x16: same K mapping, N = lane%16.
// C/D 16x16 f32: VGPR r -> M = r + 8*h, N = lane%16.
__global__ void __launch_bounds__(256)
matmul_mask_kernel(const float* __restrict__ x, const float* __restrict__ W,
                   const int* __restrict__ count, const int* __restrict__ mix,
                   float* __restrict__ out, int n) {
  const int lane  = threadIdx.x & 31;
  const int half  = lane >> 4;
  const int lm    = lane & 15;
  const int gwave = (blockIdx.x * blockDim.x + threadIdx.x) >> 5;
  const int nwav  = (gridDim.x * blockDim.x) >> 5;

  // Preload B fragments of W (30x30 row-major, zero-padded to 32x32).
  // Zero padding at k>=30 / j>=30 makes clamped A loads harmless (a * 0 == 0).
  v2f bfrag[2][8];
#pragma unroll
  for (int t = 0; t < 2; ++t) {
    const int j = 16 * t + lm;
#pragma unroll
    for (int c = 0; c < 8; ++c) {
      const int k0 = 4 * c + 2 * half;
      v2f b;
      b.x = (j < FD && k0 < FD)       ? W[k0 * FD + j]       : 0.0f;
      b.y = (j < FD && (k0 + 1) < FD) ? W[(k0 + 1) * FD + j] : 0.0f;
      bfrag[t][c] = b;
    }
  }

  const float scale = 0.18257418583505537f;  // 1/sqrt(30)
  const int ntiles = (n + 15) >> 4;

  for (int tile = gwave; tile < ntiles; tile += nwav) {
    const int base = tile << 4;
    int arow = base + lm;
    if (arow > n - 1) arow = n - 1;          // clamp, no divergent predication
    const float* xr = x + (size_t)arow * FD;

    // Phase 1: issue all 8 A-fragment loads unconditionally (straight-line b64s).
    v2f afrag[8];
#pragma unroll
    for (int c = 0; c < 8; ++c) {
      int k0 = 4 * c + 2 * half;             // pairs (0,1)..(28,29),(30,31)
      int ko = (k0 <= FD - 2) ? k0 : FD - 2; // clamp (30,31)->(28,29); B is 0 there
      afrag[c] = __builtin_nontemporal_load((const v2f*)(xr + ko)); // stream-once x
    }

    // Phase 2: 16 f32 WMMAs.
    v8f acc0 = {};
    v8f acc1 = {};
#pragma unroll
    for (int c = 0; c < 8; ++c) {
      acc0 = __builtin_amdgcn_wmma_f32_16x16x4_f32(false, afrag[c], false, bfrag[0][c],
                                                   (short)0, acc0, false, false);
      acc1 = __builtin_amdgcn_wmma_f32_16x16x4_f32(false, afrag[c], false, bfrag[1][c],
                                                   (short)0, acc1, false, false);
    }

    // Phase 3: fused criterion + window mask + scale on store.
#pragma unroll
    for (int r = 0; r < 8; ++r) {
      const int m = half * 8 + r;
      const int grow = base + m;
      if (grow < n) {
        const int cnt  = count[grow];
        const int mx   = mix[grow];
        const int crit = (cnt > DEG_THRESH ? 3 : 0) + (mx == BIGI ? 0 : mx);
        const int start = crit * 5;  // WINDOW = 5
        float* orow = out + (size_t)grow * FD;
        const int j0 = lm;
        const float v0 = acc0[r] * scale;
        __builtin_nontemporal_store((j0 >= start && j0 < start + 5) ? v0 : 0.0f,
                                    &orow[j0]);
        const int j1 = 16 + lm;
        if (j1 < FD) {
          const float v1 = acc1[r] * scale;
          __builtin_nontemporal_store((j1 >= start && j1 < start + 5) ? v1 : 0.0f,
                                      &orow[j1]);
        }
      }
    }
  }
}

extern "C" void kernel_launch(void* const* d_in, const int* in_sizes, int n_in,
                              void* d_out, int out_size, void* d_ws, size_t ws_size,
                              hipStream_t stream) {
  const float* x     = (const float*)d_in[0];
  const float* W     = (const float*)d_in[1];
  const int*   ei    = (const int*)d_in[2];
  const int*   types = (const int*)d_in[3];
  float* out = (float*)d_out;

  const int n  = in_sizes[0] / FD;  // 1,000,000 nodes
  const int ne = in_sizes[2] / 2;   // 16,000,000 edges

  int* count = (int*)d_ws;
  int* mix   = count + n;

  init_ws_kernel<<<(n + 255) / 256, 256, 0, stream>>>(count, mix, n);
  edge_kernel<<<(ne + 255) / 256, 256, 0, stream>>>(ei, types, count, mix, ne);

  const int ntiles = (n + 15) / 16;
  int blocks = (ntiles + 7) / 8;    // 8 waves (256 threads) per block
  if (blocks > 4096) blocks = 4096; // grid-stride: amortize W-fragment preload
  matmul_mask_kernel<<<blocks, 256, 0, stream>>>(x, W, count, mix, out, n);
}